// Aligner_77618648973814
// MI455X (gfx1250) — compile-verified
//
#include <hip/hip_runtime.h>
#include <math.h>

// ---- problem constants (from reference) ----
#define BB      16
#define TT      512
#define TM      2048
#define DD      256
#define HH      128
#define H3      384      // 3*H
#define NMELS   80
#define NEGV    (-1e9f)
// -0.5*log(2*pi)*NMELS
#define LOGC    (-73.51508265641551f)

typedef __attribute__((ext_vector_type(2))) float v2f;
typedef __attribute__((ext_vector_type(8))) float v8f;

// ======================================================================
// Generic fp32 WMMA GEMM:  C[M,N] = A[M,K] @ W[N,K]^T + bias[N]
// One 16x16 output tile per wave, K stepped by 4 via v_wmma_f32_16x16x4_f32.
// A-frag (16x4, M x K): lane L holds row m=L&15, k-pair base 2*(L>>4).
// B-frag (4x16, K x N): lane L holds col n=L&15, same k-pair base.
// D (16x16): vgpr r, lane L -> row 16*tm + r + 8*(L>>4), col 16*tn + (L&15).
// ======================================================================
__global__ void alg_gemm_abt_bias(const float* __restrict__ A,
                                  const float* __restrict__ W,
                                  const float* __restrict__ bias,
                                  float* __restrict__ C,
                                  int M, int N, int K)
{
    const int ntiles = N >> 4;
    const int wave = threadIdx.x >> 5;
    const int tile = blockIdx.x * (blockDim.x >> 5) + wave;
    const int tm = tile / ntiles;
    const int tn = tile % ntiles;
    if (tm * 16 >= M) return;                 // wave-uniform: EXEC stays all-1s

    const int lane = threadIdx.x & 31;
    const int half = lane >> 4;               // 0 or 1
    const int lm   = lane & 15;

    const float* arow = A + (size_t)(tm * 16 + lm) * K + 2 * half;
    const float* brow = W + (size_t)(tn * 16 + lm) * K + 2 * half;

    v8f acc = {};
    for (int k = 0; k < K; k += 4) {
        v2f a = *(const v2f*)(arow + k);
        v2f b = *(const v2f*)(brow + k);
        acc = __builtin_amdgcn_wmma_f32_16x16x4_f32(
            false, a, false, b, (short)0, acc, false, false);
    }

    const int col = tn * 16 + lm;
    const float bv = bias ? bias[col] : 0.0f;
#pragma unroll
    for (int r = 0; r < 8; ++r) {
        const int row = tm * 16 + r + 8 * half;
        C[(size_t)row * N + col] = acc[r] + bv;
    }
}

// ======================================================================
// log_prior GEMM (per batch):
//   logp[b][m][t] = mel[b,m,:].xh[b,t,:] - 0.5||xh||^2 - 0.5||mel||^2 + LOGC
//   masked to NEGV outside attn_mask.
// A = mel[b] (M=TM x K=80), B = xh[b]^T (K=80 x N=TT).
// ======================================================================
__global__ void alg_logprior_gemm(const float* __restrict__ mel,
                                  const float* __restrict__ xh,
                                  const float* __restrict__ sqm,
                                  const float* __restrict__ sqx,
                                  const unsigned char* __restrict__ tmask,
                                  const unsigned char* __restrict__ mmask,
                                  float* __restrict__ logp)
{
    const int b = blockIdx.y;
    const int ntiles = TT >> 4;               // 32
    const int wave = threadIdx.x >> 5;
    const int tile = blockIdx.x * (blockDim.x >> 5) + wave;
    const int tm = tile / ntiles;             // mel tile   (0..127)
    const int tn = tile % ntiles;             // text tile  (0..31)

    const int lane = threadIdx.x & 31;
    const int half = lane >> 4;
    const int lm   = lane & 15;

    const float* arow = mel + ((size_t)b * TM + tm * 16 + lm) * NMELS + 2 * half;
    const float* brow = xh  + ((size_t)b * TT + tn * 16 + lm) * NMELS + 2 * half;

    v8f acc = {};
#pragma unroll
    for (int k = 0; k < NMELS; k += 4) {      // 20 wmma steps
        v2f a = *(const v2f*)(arow + k);
        v2f b2 = *(const v2f*)(brow + k);
        acc = __builtin_amdgcn_wmma_f32_16x16x4_f32(
            false, a, false, b2, (short)0, acc, false, false);
    }

    const int col = tn * 16 + lm;             // t
    const float nsx = -0.5f * sqx[b * TT + col];
    const bool tok = tmask[b * TT + col] != 0;
#pragma unroll
    for (int r = 0; r < 8; ++r) {
        const int row = tm * 16 + r + 8 * half;   // m
        float v = acc[r] + nsx - 0.5f * sqm[b * TM + row] + LOGC;
        const bool ok = tok && (mmask[b * TM + row] != 0);
        logp[((size_t)b * TM + row) * TT + col] = ok ? v : NEGV;
    }
}

// ======================================================================
// Embedding lookup: x[bt, d] = emb[text[bt], d]
// ======================================================================
__global__ void alg_embed(const int* __restrict__ text,
                          const float* __restrict__ emb,
                          float* __restrict__ x)
{
    const int i = blockIdx.x * 256 + threadIdx.x;   // over B*TT*D
    const int bt = i >> 8;
    const int d  = i & 255;
    x[i] = emb[text[bt] * DD + d];
}

// residual + concat: x[bt, d] += (d<H ? hs_f[bt,d] : hs_b[bt,d-H])
__global__ void alg_combine(const float* __restrict__ hs_f,
                            const float* __restrict__ hs_b,
                            float* __restrict__ x)
{
    const int i = blockIdx.x * 256 + threadIdx.x;
    const int bt = i >> 8;
    const int d  = i & 255;
    const float add = (d < HH) ? hs_f[bt * HH + d] : hs_b[bt * HH + (d - HH)];
    x[i] += add;
}

// ======================================================================
// GRU recurrence, one workgroup per batch element. 384 threads:
//   thread j computes hg[j] = b_hh[j] + w_hh[j,:].h  (w_hh row held in VGPRs)
//   threads 0..127 then do the torch-GRU gate math and update h in LDS.
// ======================================================================
__global__ void __launch_bounds__(H3) alg_gru_scan(
        const float* __restrict__ xg,     // [B,TT,3H] precomputed input gates
        const float* __restrict__ w_hh,   // [3H,H]
        const float* __restrict__ b_hh,   // [3H]
        float* __restrict__ hs,           // [B,TT,H]
        int reverse)
{
    __shared__ float h_lds[HH];
    __shared__ float hg_lds[H3];

    const int b = blockIdx.x;
    const int j = threadIdx.x;            // 0..383

    float w[HH];
#pragma unroll
    for (int k = 0; k < HH; ++k) w[k] = w_hh[j * HH + k];
    const float bj = b_hh[j];
    if (j < HH) h_lds[j] = 0.0f;
    __syncthreads();

    for (int s = 0; s < TT; ++s) {
        const int t = reverse ? (TT - 1 - s) : s;
        float acc = bj;
#pragma unroll
        for (int k = 0; k < HH; ++k) acc += w[k] * h_lds[k];
        hg_lds[j] = acc;
        __syncthreads();
        if (j < HH) {
            const float* xgt = xg + ((size_t)b * TT + t) * H3;
            const float xr = xgt[j], xz = xgt[j + HH], xn = xgt[j + 2 * HH];
            const float hr = hg_lds[j], hz = hg_lds[j + HH], hn = hg_lds[j + 2 * HH];
            const float r = 1.0f / (1.0f + __expf(-(xr + hr)));
            const float z = 1.0f / (1.0f + __expf(-(xz + hz)));
            const float n = tanhf(xn + r * hn);
            const float hnew = (1.0f - z) * n + z * h_lds[j];
            h_lds[j] = hnew;
            hs[((size_t)b * TT + t) * HH + j] = hnew;
        }
        __syncthreads();
    }
}

// row-wise squared norms over NMELS columns
__global__ void alg_rowsq(const float* __restrict__ a,
                          float* __restrict__ out, int rows)
{
    const int i = blockIdx.x * blockDim.x + threadIdx.x;
    if (i >= rows) return;
    const float* r = a + (size_t)i * NMELS;
    float s = 0.0f;
#pragma unroll 8
    for (int n = 0; n < NMELS; ++n) s += r[n] * r[n];
    out[i] = s;
}

// mask sums -> lens[0..B-1]=tx_len, lens[B..2B-1]=ty_len
__global__ void alg_lengths(const unsigned char* __restrict__ tmask,
                            const unsigned char* __restrict__ mmask,
                            int* __restrict__ lens)
{
    const int b = threadIdx.x;
    if (b >= BB) return;
    int s = 0;
    for (int t = 0; t < TT; ++t) s += (tmask[b * TT + t] != 0);
    lens[b] = s;
    int s2 = 0;
    for (int m = 0; m < TM; ++m) s2 += (mmask[b * TM + m] != 0);
    lens[BB + b] = s2;
}

// ======================================================================
// MAS forward DP: one workgroup per batch, thread x owns q[x].
// Stores only the 1-byte "diagonal move" flag dir[y][x] = Q[y-1][x] < Q[y-1][x-1].
// ======================================================================
__global__ void __launch_bounds__(TT) alg_mas_forward(
        const float* __restrict__ logp,     // [B][TM][TT]
        const int* __restrict__ lens,
        unsigned char* __restrict__ dir)    // [B][TM][TT]
{
    __shared__ float qs[TT];
    const int b = blockIdx.x;
    const int x = threadIdx.x;              // 0..511
    const int txl = lens[b];
    const int tyl = lens[BB + b];
    const float* lp = logp + (size_t)b * TM * TT;
    unsigned char* db = dir + (size_t)b * TM * TT;

    float q = NEGV;
    for (int y = 0; y < TM; ++y) {
        qs[x] = q;
        __syncthreads();
        const float qshift = (x > 0) ? qs[x - 1] : NEGV;
        db[(size_t)y * TT + x] = (q < qshift) ? 1 : 0;
        const float v = lp[(size_t)y * TT + x];
        float qn = v + fmaxf(q, qshift);
        if (y == 0) qn = (x == 0) ? v : NEGV;
        const int lo = txl + y - tyl;
        const bool valid = (x <= y) && (x >= lo) && (x < txl);
        q = valid ? qn : NEGV;
        __syncthreads();
    }
}

// backtrack: one thread per batch, emit chosen text index per mel frame
__global__ void alg_mas_backtrack(const unsigned char* __restrict__ dir,
                                  const int* __restrict__ lens,
                                  int* __restrict__ idx)
{
    const int b = blockIdx.x * blockDim.x + threadIdx.x;
    if (b >= BB) return;
    const int txl = lens[b];
    const int tyl = lens[BB + b];
    const unsigned char* db = dir + (size_t)b * TM * TT;
    int index = txl - 1;
    for (int y = TM - 1; y >= 0; --y) {
        const bool active = (y < tyl) && (index >= 0);
        idx[b * TM + y] = active ? index : -1;
        if (active) {
            const bool move =
                ((index == y) || (db[(size_t)y * TT + index] != 0)) && (index != 0);
            if (move) --index;
        }
    }
}

// out[b,m,:] = xh[b, idx[b,m], :]  (zeros outside valid mel frames)
__global__ void alg_gather_out(const float* __restrict__ xh,
                               const int* __restrict__ idx,
                               const int* __restrict__ lens,
                               float* __restrict__ out)
{
    const int bm = blockIdx.x;
    const int n = threadIdx.x;
    if (n >= NMELS) return;
    const int b = bm / TM;
    const int m = bm % TM;
    const int t = idx[b * TM + m];
    float v = 0.0f;
    if (m < lens[BB + b] && t >= 0)
        v = xh[((size_t)b * TT + t) * NMELS + n];
    out[(size_t)bm * NMELS + n] = v;
}

// ======================================================================
extern "C" void kernel_launch(void* const* d_in, const int* in_sizes, int n_in,
                              void* d_out, int out_size, void* d_ws, size_t ws_size,
                              hipStream_t stream) {
    const int*           text   = (const int*)d_in[0];
    const unsigned char* tmask  = (const unsigned char*)d_in[1];
    const float*         mel    = (const float*)d_in[2];
    const unsigned char* mmask  = (const unsigned char*)d_in[3];
    const float*         emb    = (const float*)d_in[4];
    const float*         w_ih   = (const float*)d_in[5];   // [2,2,384,256]
    const float*         w_hh   = (const float*)d_in[6];   // [2,2,384,128]
    const float*         b_ih   = (const float*)d_in[7];   // [2,2,384]
    const float*         b_hh   = (const float*)d_in[8];   // [2,2,384]
    const float*         head_w = (const float*)d_in[9];   // [80,256]
    const float*         head_b = (const float*)d_in[10];  // [80]
    float*               out    = (float*)d_out;

    char* ws = (char*)d_ws;
    size_t off = 0;
    auto alloc = [&](size_t bytes) -> void* {
        void* p = ws + off;
        off += (bytes + 255) & ~(size_t)255;
        return p;
    };

    float* x    = (float*)alloc((size_t)BB * TT * DD * 4);       // 8 MB
    float* xg_f = (float*)alloc((size_t)BB * TT * H3 * 4);       // 12 MB
    float* xg_b = (float*)alloc((size_t)BB * TT * H3 * 4);       // 12 MB
    float* hs_f = (float*)alloc((size_t)BB * TT * HH * 4);       // 4 MB
    float* hs_b = (float*)alloc((size_t)BB * TT * HH * 4);       // 4 MB
    float* xh   = (float*)alloc((size_t)BB * TT * NMELS * 4);    // 2.5 MB
    float* sqx  = (float*)alloc((size_t)BB * TT * 4);
    float* sqm  = (float*)alloc((size_t)BB * TM * 4);
    int*   lens = (int*)  alloc((size_t)2 * BB * 4);
    int*   idx  = (int*)  alloc((size_t)BB * TM * 4);
    float* logp = (float*)alloc((size_t)BB * TM * TT * 4);       // 64 MB
    unsigned char* dirb = (unsigned char*)alloc((size_t)BB * TM * TT); // 16 MB

    alg_lengths<<<1, 32, 0, stream>>>(tmask, mmask, lens);
    alg_embed<<<(BB * TT * DD) / 256, 256, 0, stream>>>(text, emb, x);

    for (int l = 0; l < 2; ++l) {
        const float* wif = w_ih + (size_t)(l * 2 + 0) * H3 * DD;
        const float* wib = w_ih + (size_t)(l * 2 + 1) * H3 * DD;
        const float* whf = w_hh + (size_t)(l * 2 + 0) * H3 * HH;
        const float* whb = w_hh + (size_t)(l * 2 + 1) * H3 * HH;
        const float* bif = b_ih + (size_t)(l * 2 + 0) * H3;
        const float* bib = b_ih + (size_t)(l * 2 + 1) * H3;
        const float* bhf = b_hh + (size_t)(l * 2 + 0) * H3;
        const float* bhb = b_hh + (size_t)(l * 2 + 1) * H3;

        // input-gate GEMMs: [8192,256] @ [256,384] (+bias) -> 512*24 tiles
        const int g1 = (BB * TT / 16) * (H3 / 16) / 4;
        alg_gemm_abt_bias<<<g1, 128, 0, stream>>>(x, wif, bif, xg_f, BB * TT, H3, DD);
        alg_gemm_abt_bias<<<g1, 128, 0, stream>>>(x, wib, bib, xg_b, BB * TT, H3, DD);

        // sequential recurrences
        alg_gru_scan<<<BB, H3, 0, stream>>>(xg_f, whf, bhf, hs_f, 0);
        alg_gru_scan<<<BB, H3, 0, stream>>>(xg_b, whb, bhb, hs_b, 1);

        alg_combine<<<(BB * TT * DD) / 256, 256, 0, stream>>>(hs_f, hs_b, x);
    }

    // head projection: [8192,256] @ [256,80] + bias -> 512*5 tiles
    alg_gemm_abt_bias<<<(BB * TT / 16) * (NMELS / 16) / 4, 128, 0, stream>>>(
        x, head_w, head_b, xh, BB * TT, NMELS, DD);

    alg_rowsq<<<(BB * TT + 255) / 256, 256, 0, stream>>>(xh, sqx, BB * TT);
    alg_rowsq<<<(BB * TM + 255) / 256, 256, 0, stream>>>(mel, sqm, BB * TM);

    // log_prior: per-batch [2048,80] @ [80,512] with norm/mask epilogue
    dim3 glp((TM / 16) * (TT / 16) / 4, BB);
    alg_logprior_gemm<<<glp, 128, 0, stream>>>(mel, xh, sqm, sqx, tmask, mmask, logp);

    alg_mas_forward<<<BB, TT, 0, stream>>>(logp, lens, dirb);
    alg_mas_backtrack<<<1, 32, 0, stream>>>(dirb, lens, idx);
    alg_gather_out<<<BB * TM, 128, 0, stream>>>(xh, idx, lens, out);
}